// ScaledDotProductAttentionNonSelf_1692217115126
// MI455X (gfx1250) — compile-verified
//
#include <hip/hip_runtime.h>

typedef __attribute__((ext_vector_type(2))) float v2f;
typedef __attribute__((ext_vector_type(8))) float v8f;

constexpr int NH = 8;    // heads
constexpr int NB = 256;  // batch
constexpr int SL = 128;  // sequence length (Lq == Lk)
constexpr int HD = 64;   // head dim

constexpr int QK_STRIDE = 68;   // padded LDS row stride for Q,K (floats)
constexpr int VT_STRIDE = 132;  // padded LDS row stride for V^T, P (floats)

// LDS layout (floats):
//  [0            , 128*68      )  Qs   -- reused as Ps (128*132 = 16896 <= 17408) after barrier
//  [128*68       , 2*128*68    )  Ks
//  [2*128*68     , +64*132     )  Vt
constexpr int LDS_FLOATS = 2 * SL * QK_STRIDE + HD * VT_STRIDE;  // 25856 floats = 103424 B

// Low 32 bits of a generic pointer into LDS == byte offset within the wave's
// LDS allocation.
__device__ __forceinline__ unsigned int lds_off(const void* p) {
    return (unsigned int)(unsigned long long)p;
}

__global__ __launch_bounds__(256)
void attn_fused_kernel(const float* __restrict__ gq, const float* __restrict__ gk,
                       const float* __restrict__ gv, const float* __restrict__ gbias,
                       float* __restrict__ gout, float* __restrict__ gattn)
{
    extern __shared__ float smem[];
    float* Qs  = smem;
    float* Ks  = smem + SL * QK_STRIDE;
    float* Ps  = smem;                       // overlays Qs/Ks after all K reads done
    float* Vts = smem + 2 * SL * QK_STRIDE;

    const int tid   = threadIdx.x;
    const int lane  = tid & 31;
    const int wave  = tid >> 5;      // 8 waves: M-tile id
    const int laneN = lane & 15;
    const int laneH = lane >> 4;     // half-wave select (K-pair / M-upper-half)

    const int b = blockIdx.x;
    const int h = blockIdx.y;
    const size_t base = ((size_t)h * NB + b) * (SL * HD);

    // ---- stage Q,K via async global->LDS DMA (no VGPR round trip) ----
    {
        const float* qp = gq + base;
        const float* kp = gk + base;
#pragma unroll
        for (int i = 0; i < 8; ++i) {
            const int f   = tid + 256 * i;        // float4 index over 128x64
            const int row = f >> 4;
            const int c4  = (f & 15) * 4;
            const unsigned int lq = lds_off(Qs + row * QK_STRIDE + c4);
            const unsigned int lk = lds_off(Ks + row * QK_STRIDE + c4);
            const float* gpq = qp + (size_t)f * 4;
            const float* gpk = kp + (size_t)f * 4;
            asm volatile("global_load_async_to_lds_b128 %0, %1, off"
                         :: "v"(lq), "v"(gpq) : "memory");
            asm volatile("global_load_async_to_lds_b128 %0, %1, off"
                         :: "v"(lk), "v"(gpk) : "memory");
        }
    }

    // ---- V transposed into LDS through VGPRs (async can't transpose) ----
    {
        const float4* v4 = (const float4*)(gv + base);
#pragma unroll
        for (int i = 0; i < 8; ++i) {
            const int f   = tid + 256 * i;
            const int row = f >> 4;
            const int c4  = (f & 15) * 4;
            float4 vv = v4[f];
            Vts[(c4 + 0) * VT_STRIDE + row] = vv.x;
            Vts[(c4 + 1) * VT_STRIDE + row] = vv.y;
            Vts[(c4 + 2) * VT_STRIDE + row] = vv.z;
            Vts[(c4 + 3) * VT_STRIDE + row] = vv.w;
        }
    }

    // ---- warm L2/L0 with this wave's 16 bias rows while WMMAs run ----
    const int mrow0 = wave * 16 + laneH * 8;
    const float* biasH = gbias + (size_t)h * SL * SL;
#pragma unroll
    for (int r = 0; r < 8; ++r)
        __builtin_prefetch(biasH + (size_t)(mrow0 + r) * SL + laneN * 8, 0, 3);

    asm volatile("s_wait_asynccnt 0x0" ::: "memory");
    __syncthreads();

    // ---- S = Q K^T : wave's 16x128 block as 8 WMMA accumulators ----
    v8f acc[8];
#pragma unroll
    for (int nt = 0; nt < 8; ++nt)
#pragma unroll
        for (int j = 0; j < 8; ++j) acc[nt][j] = 0.0f;

    const int mrowA = wave * 16 + laneN;  // A-fragment row for this lane
#pragma unroll 4
    for (int ks = 0; ks < 16; ++ks) {     // K dim 64, 4 per WMMA
        const int c = 4 * ks + 2 * laneH;
        v2f a = *(const v2f*)(Qs + mrowA * QK_STRIDE + c);
#pragma unroll
        for (int nt = 0; nt < 8; ++nt) {
            v2f bb = *(const v2f*)(Ks + (nt * 16 + laneN) * QK_STRIDE + c);
            acc[nt] = __builtin_amdgcn_wmma_f32_16x16x4_f32(
                false, a, false, bb, (short)0, acc[nt], false, false);
        }
    }

    // ---- scale + bias + softmax along Lk (row = 16-lane half x 8 tiles) ----
    float* attnO = gattn + ((size_t)h * NB + b) * (SL * SL);
#pragma unroll
    for (int r = 0; r < 8; ++r) {
        const int m = mrow0 + r;
        const float* brow = biasH + (size_t)m * SL + laneN;
        float s[8];
        float mx = -3.0e38f;
#pragma unroll
        for (int nt = 0; nt < 8; ++nt) {
            s[nt] = acc[nt][r] * 0.125f + brow[nt * 16];  // /temperature + pos_bias
            mx = fmaxf(mx, s[nt]);
        }
#pragma unroll
        for (int msk = 1; msk <= 8; msk <<= 1)
            mx = fmaxf(mx, __shfl_xor(mx, msk, 32));
        float sum = 0.0f;
#pragma unroll
        for (int nt = 0; nt < 8; ++nt) { s[nt] = __expf(s[nt] - mx); sum += s[nt]; }
#pragma unroll
        for (int msk = 1; msk <= 8; msk <<= 1)
            sum += __shfl_xor(sum, msk, 32);
        const float inv = 1.0f / sum;
        float* arow = attnO + (size_t)m * SL + laneN;
#pragma unroll
        for (int nt = 0; nt < 8; ++nt) {
            const float p = s[nt] * inv;
            acc[nt][r] = p;       // keep P for the PV matmul
            arow[nt * 16] = p;    // attn output
        }
    }

    // all waves must be finished reading Qs/Ks before P overwrites that region
    __syncthreads();

    // C-layout -> A-layout transpose via LDS (each wave touches only its own rows)
#pragma unroll
    for (int r = 0; r < 8; ++r) {
        const int m = mrow0 + r;
#pragma unroll
        for (int nt = 0; nt < 8; ++nt)
            Ps[m * VT_STRIDE + nt * 16 + laneN] = acc[nt][r];
    }

    // ---- O = P V : 16x64 per wave ----
    v8f od[4];
#pragma unroll
    for (int dt = 0; dt < 4; ++dt)
#pragma unroll
        for (int j = 0; j < 8; ++j) od[dt][j] = 0.0f;

#pragma unroll 4
    for (int ks = 0; ks < 32; ++ks) {     // K dim 128, 4 per WMMA
        const int c = 4 * ks + 2 * laneH;
        v2f a = *(const v2f*)(Ps + mrowA * VT_STRIDE + c);
#pragma unroll
        for (int dt = 0; dt < 4; ++dt) {
            v2f bb = *(const v2f*)(Vts + (dt * 16 + laneN) * VT_STRIDE + c);
            od[dt] = __builtin_amdgcn_wmma_f32_16x16x4_f32(
                false, a, false, bb, (short)0, od[dt], false, false);
        }
    }

    float* outO = gout + base;
#pragma unroll
    for (int r = 0; r < 8; ++r) {
        const int m = mrow0 + r;
#pragma unroll
        for (int dt = 0; dt < 4; ++dt)
            outO[(size_t)m * HD + dt * 16 + laneN] = od[dt][r];
    }
}

extern "C" void kernel_launch(void* const* d_in, const int* in_sizes, int n_in,
                              void* d_out, int out_size, void* d_ws, size_t ws_size,
                              hipStream_t stream) {
    const float* q    = (const float*)d_in[0];
    const float* k    = (const float*)d_in[1];
    const float* v    = (const float*)d_in[2];
    const float* bias = (const float*)d_in[3];

    float* out  = (float*)d_out;
    float* attn = (float*)d_out + (size_t)NH * NB * SL * HD;  // tuple order: (out, attn)

    const int ldsBytes = LDS_FLOATS * (int)sizeof(float);     // 103424 B (>64KB -> opt in)
    (void)hipFuncSetAttribute((const void*)attn_fused_kernel,
                              hipFuncAttributeMaxDynamicSharedMemorySize, ldsBytes);

    dim3 grid(NB, NH);  // one workgroup per (h, b)
    attn_fused_kernel<<<grid, 256, ldsBytes, stream>>>(q, k, v, bias, out, attn);
}